// SubGraph_33792802685128
// MI455X (gfx1250) — compile-verified
//
#include <hip/hip_runtime.h>
#include <hip/hip_bf16.h>
#include <math.h>

#define N_NODES 100000
#define N_EDGES 800000
#define NCLUST  10000
#define LN_EPS  1e-5f

typedef __attribute__((ext_vector_type(2))) float v2f;
typedef __attribute__((ext_vector_type(8))) float v8f;

// ---- monotonic fp32 <-> u32 encoding so u32 atomicMax == float max ----
__device__ __forceinline__ unsigned enc_f32(float f) {
    unsigned u = __float_as_uint(f);
    return (u & 0x80000000u) ? ~u : (u | 0x80000000u);
}
__device__ __forceinline__ float dec_f32(unsigned e) {
    unsigned u = (e & 0x80000000u) ? (e & 0x7FFFFFFFu) : ~e;
    return __uint_as_float(u);
}
#define ENC_NEGINF 0x007FFFFFu   // enc(-inf)

// ---------------------------------------------------------------------
// Y[M x Ncols] = X[M x K] @ W[K x Ncols] + bias, via V_WMMA_F32_16X16X4_F32.
// grid.x = M/16 row tiles, grid.y = Ncols/64 col super-tiles, 4 waves/block,
// wave w owns cols [64*by + 16*w, +16). EXEC all-ones everywhere (exact tiling).
// ---------------------------------------------------------------------
__global__ __launch_bounds__(128)
void wmma_gemm_bias(const float* __restrict__ X, int ldx,
                    const float* __restrict__ W, int ldw,
                    const float* __restrict__ bias,
                    float* __restrict__ Y, int ldy, int K)
{
    const int wave = threadIdx.x >> 5;
    const int lane = threadIdx.x & 31;
    const int hl = lane >> 4;        // half-wave: 0 or 1
    const int ll = lane & 15;
    const int row0 = blockIdx.x << 4;
    const int col0 = (blockIdx.y << 6) + (wave << 4);

    // A frag source: lane(l<16) -> row row0+l, K = {k, k+1}; lane(l>=16) -> K = {k+2, k+3}
    const float* xr = X + (size_t)(row0 + ll) * ldx + 2 * hl;
    // B frag source: vgpr0 = W[k + 2*hl][col], vgpr1 = W[k+1 + 2*hl][col]
    const float* wc = W + (size_t)(2 * hl) * ldw + col0 + ll;

    v8f acc = {0.f, 0.f, 0.f, 0.f, 0.f, 0.f, 0.f, 0.f};
    for (int k = 0; k < K; k += 4) {
        v2f a, b;
        a.x = xr[k];
        a.y = xr[k + 1];
        b.x = wc[(size_t)k * ldw];
        b.y = wc[(size_t)(k + 1) * ldw];
        acc = __builtin_amdgcn_wmma_f32_16x16x4_f32(
            /*neg_a=*/false, a, /*neg_b=*/false, b,
            /*c_mod=*/(short)0, acc, /*reuse_a=*/false, /*reuse_b=*/false);
    }

    const float bv = bias[col0 + ll];
    // C/D layout: vgpr i holds row (row0 + i + 8*hl), col = col0 + ll
    float* yr = Y + (size_t)(row0 + 8 * hl) * ldy + col0 + ll;
#pragma unroll
    for (int i = 0; i < 8; ++i) yr[(size_t)i * ldy] = acc[i] + bv;
}

// ---------------------------------------------------------------------
// Row-wise LayerNorm(64) + ReLU, in place. One wave per row, 2 channels/lane.
// ---------------------------------------------------------------------
__global__ __launch_bounds__(256)
void ln_relu(float* __restrict__ h, const float* __restrict__ g,
             const float* __restrict__ be)
{
    const int row  = blockIdx.x * 8 + (threadIdx.x >> 5);
    const int lane = threadIdx.x & 31;
    float* hr = h + (size_t)row * 64;
    float v0 = hr[lane], v1 = hr[lane + 32];

    float s = v0 + v1;
    for (int m = 16; m; m >>= 1) s += __shfl_xor(s, m, 32);
    const float mu = s * (1.f / 64.f);
    const float d0 = v0 - mu, d1 = v1 - mu;
    float q = d0 * d0 + d1 * d1;
    for (int m = 16; m; m >>= 1) q += __shfl_xor(q, m, 32);
    const float r = rsqrtf(q * (1.f / 64.f) + LN_EPS);

    float y0 = d0 * r * g[lane] + be[lane];
    float y1 = d1 * r * g[lane + 32] + be[lane + 32];
    hr[lane]      = y0 > 0.f ? y0 : 0.f;
    hr[lane + 32] = y1 > 0.f ? y1 : 0.f;
}

// ---- aggr half of concat buffer: init to enc(-inf) ----
__global__ void init_aggr(unsigned* __restrict__ buf, int C, int ld, int nrows)
{
    int i = blockIdx.x * blockDim.x + threadIdx.x;
    int row = i / C, c = i - row * C;
    if (row < nrows) buf[(size_t)row * ld + C + c] = ENC_NEGINF;
}

// ---- per-edge gather of h[src], u32-encoded atomic max into aggr[dst] ----
__global__ void edge_scatter_max(float* __restrict__ xcat,
                                 const int* __restrict__ ei, int C, int ld)
{
    int gid = blockIdx.x * blockDim.x + threadIdx.x;
    int cg = C >> 2;
    int e = gid / cg;
    if (e >= N_EDGES) return;
    int c = (gid - e * cg) << 2;
    int src = ei[e];
    int dst = ei[N_EDGES + e];
    const float4 v = *(const float4*)(xcat + (size_t)src * ld + c);
    unsigned* a = (unsigned*)xcat + (size_t)dst * ld + C + c;
    atomicMax(a + 0, enc_f32(v.x));
    atomicMax(a + 1, enc_f32(v.y));
    atomicMax(a + 2, enc_f32(v.z));
    atomicMax(a + 3, enc_f32(v.w));
}

// ---- decode aggr half back to fp32; nodes with no in-edges (-inf) -> 0 ----
__global__ void decode_aggr(unsigned* __restrict__ buf, int C, int ld, int nrows)
{
    int i = blockIdx.x * blockDim.x + threadIdx.x;
    int row = i / C, c = i - row * C;
    if (row >= nrows) return;
    unsigned* p = buf + (size_t)row * ld + C + c;
    float f = dec_f32(*p);
    if (!isfinite(f)) f = 0.f;
    *(float*)p = f;
}

// ---- cluster max-pool: init, scatter, decode ----
__global__ void fill_out(unsigned* __restrict__ out)
{
    int i = blockIdx.x * blockDim.x + threadIdx.x;
    if (i < NCLUST * 512) out[i] = ENC_NEGINF;
}

__global__ void pool_scatter_max(const float* __restrict__ x3,
                                 const int* __restrict__ cluster,
                                 unsigned* __restrict__ out)
{
    int gid = blockIdx.x * blockDim.x + threadIdx.x;
    int row = gid >> 7;            // 128 float4-groups per row of 512
    if (row >= N_NODES) return;
    int c = (gid & 127) << 2;
    int cl = cluster[row];
    const float4 v = *(const float4*)(x3 + (size_t)row * 512 + c);
    unsigned* a = out + (size_t)cl * 512 + c;
    atomicMax(a + 0, enc_f32(v.x));
    atomicMax(a + 1, enc_f32(v.y));
    atomicMax(a + 2, enc_f32(v.z));
    atomicMax(a + 3, enc_f32(v.w));
}

__global__ void decode_out(unsigned* __restrict__ out)
{
    int i = blockIdx.x * blockDim.x + threadIdx.x;
    if (i >= NCLUST * 512) return;
    float f = dec_f32(out[i]);
    if (!isfinite(f)) f = 0.f;
    ((float*)out)[i] = f;
}

// ---- column-wise squared L2 norms (one block per column) ----
__global__ __launch_bounds__(256)
void col_norm2(const float* __restrict__ out, float* __restrict__ norms)
{
    __shared__ float red[256];
    const int c = blockIdx.x;
    float s = 0.f;
    for (int r = threadIdx.x; r < NCLUST; r += 256) {
        float v = out[(size_t)r * 512 + c];
        s += v * v;
    }
    red[threadIdx.x] = s;
    __syncthreads();
    for (int off = 128; off > 0; off >>= 1) {
        if (threadIdx.x < off) red[threadIdx.x] += red[threadIdx.x + off];
        __syncthreads();
    }
    if (threadIdx.x == 0) norms[c] = red[0];
}

__global__ void col_div(float* __restrict__ out, const float* __restrict__ norms)
{
    int i = blockIdx.x * blockDim.x + threadIdx.x;
    if (i < NCLUST * 512) out[i] *= rsqrtf(norms[i & 511]);
}

// ---------------------------------------------------------------------
extern "C" void kernel_launch(void* const* d_in, const int* in_sizes, int n_in,
                              void* d_out, int out_size, void* d_ws, size_t ws_size,
                              hipStream_t stream)
{
    const float* x  = (const float*)d_in[0];
    const int*   ei = (const int*)d_in[1];
    const int*   cl = (const int*)d_in[2];
    float* out = (float*)d_out;

    float* ws    = (float*)d_ws;
    float* hmid  = ws;                                   // N * 64
    float* buf0  = hmid + (size_t)N_NODES * 64;          // N * 512 (x1 then x3)
    float* buf1  = buf0 + (size_t)N_NODES * 512;         // N * 256 (x2)
    float* norms = buf1 + (size_t)N_NODES * 256;         // 512

    const float* cur = x;
    int C = 64;
    float* outs[3] = {buf0, buf1, buf0};

    for (int i = 0; i < 3; ++i) {
        const float* W1 = (const float*)d_in[3 + 6 * i];
        const float* b1 = (const float*)d_in[4 + 6 * i];
        const float* g  = (const float*)d_in[5 + 6 * i];
        const float* be = (const float*)d_in[6 + 6 * i];
        const float* W2 = (const float*)d_in[7 + 6 * i];
        const float* b2 = (const float*)d_in[8 + 6 * i];
        float* xo = outs[i];
        const int ld = 2 * C;

        // hmid = cur @ W1 + b1   [N x 64]
        wmma_gemm_bias<<<dim3(N_NODES / 16, 1), 128, 0, stream>>>(
            cur, C, W1, 64, b1, hmid, 64, C);
        // LayerNorm + ReLU in place
        ln_relu<<<N_NODES / 8, 256, 0, stream>>>(hmid, g, be);
        // h = hmid @ W2 + b2, written into concat layout [h | aggr], stride 2C
        wmma_gemm_bias<<<dim3(N_NODES / 16, C / 64), 128, 0, stream>>>(
            hmid, 64, W2, C, b2, xo, ld, 64);

        const int tot = N_NODES * C;
        init_aggr<<<(tot + 255) / 256, 256, 0, stream>>>((unsigned*)xo, C, ld, N_NODES);
        const long et = (long)N_EDGES * (C / 4);
        edge_scatter_max<<<(unsigned)((et + 255) / 256), 256, 0, stream>>>(xo, ei, C, ld);
        decode_aggr<<<(tot + 255) / 256, 256, 0, stream>>>((unsigned*)xo, C, ld, N_NODES);

        cur = xo;
        C = ld;
    }

    // cur = x3 [N x 512] in buf0
    fill_out<<<(NCLUST * 512 + 255) / 256, 256, 0, stream>>>((unsigned*)out);
    pool_scatter_max<<<(N_NODES * 128 + 255) / 256, 256, 0, stream>>>(
        cur, cl, (unsigned*)out);
    decode_out<<<(NCLUST * 512 + 255) / 256, 256, 0, stream>>>((unsigned*)out);
    col_norm2<<<512, 256, 0, stream>>>(out, norms);
    col_div<<<(NCLUST * 512 + 255) / 256, 256, 0, stream>>>(out, norms);
}